// MultiLabelGNN_68831145886538
// MI455X (gfx1250) — compile-verified
//
#include <hip/hip_runtime.h>
#include <hip/hip_bf16.h>
#include <math.h>

typedef __attribute__((ext_vector_type(2))) float v2f;
typedef __attribute__((ext_vector_type(8))) float v8f;

// ---------------------------------------------------------------------------
// WMMA fp32 GEMM: C[M,N] = A[M,K] @ B[K,N] (+ C if accumulate) (+ bias[N])
// One wave computes a 16x64 tile (4 x v8f accumulators), K stepped by 4 with
// V_WMMA_F32_16X16X4_F32; the A fragment is reused across the 4 N-subtiles.
// M multiple of 16, N multiple of 64, K multiple of 4.
// A frag  (16x4):  lane m = lane&15, VGPR0/1 = K = (lane>>4)*2 + {0,1}
// B frag  (4x16):  lane n = lane&15, VGPR0/1 = K = (lane>>4)*2 + {0,1}
// C/D     (16x16): lane n = lane&15, VGPR i  = M = i + (lane>>4)*8
// ---------------------------------------------------------------------------
__global__ __launch_bounds__(32)
void k_gemm_wmma(const float* __restrict__ A, const float* __restrict__ B,
                 float* __restrict__ C, int M, int N, int K,
                 const float* __restrict__ bias, int accumulate) {
  const int lane = threadIdx.x & 31;
  const int bn0  = blockIdx.x * 64;
  const int bm0  = blockIdx.y * 16;
  const int nn   = lane & 15;
  const int half = lane >> 4;
  const int kb   = half * 2;

  v8f acc[4];
  if (accumulate) {
#pragma unroll
    for (int t = 0; t < 4; ++t)
#pragma unroll
      for (int i = 0; i < 8; ++i)
        acc[t][i] = C[(size_t)(bm0 + i + half * 8) * N + bn0 + t * 16 + nn];
  } else {
#pragma unroll
    for (int t = 0; t < 4; ++t)
#pragma unroll
      for (int i = 0; i < 8; ++i) acc[t][i] = 0.f;
  }

  const float* arow = A + (size_t)(bm0 + nn) * K;
  for (int k0 = 0; k0 < K; k0 += 4) {
    v2f a;
    a.x = arow[k0 + kb];
    a.y = arow[k0 + kb + 1];
    const float* brow0 = B + (size_t)(k0 + kb) * N + bn0 + nn;
    const float* brow1 = brow0 + N;
#pragma unroll
    for (int t = 0; t < 4; ++t) {
      v2f b;
      b.x = brow0[t * 16];
      b.y = brow1[t * 16];
      acc[t] = __builtin_amdgcn_wmma_f32_16x16x4_f32(
          /*neg_a=*/false, a, /*neg_b=*/false, b,
          /*c_mod=*/(short)0, acc[t], /*reuse_a=*/false, /*reuse_b=*/false);
    }
  }

#pragma unroll
  for (int t = 0; t < 4; ++t) {
    const float bv = bias ? bias[bn0 + t * 16 + nn] : 0.f;
#pragma unroll
    for (int i = 0; i < 8; ++i)
      C[(size_t)(bm0 + i + half * 8) * N + bn0 + t * 16 + nn] = acc[t][i] + bv;
  }
}

// ---------------------------------------------------------------------------
__global__ void k_fill(float* __restrict__ p, float v, size_t n) {
  size_t i = (size_t)blockIdx.x * blockDim.x + threadIdx.x;
  if (i < n) p[i] = v;
}

__device__ __forceinline__ void atomicMaxF(float* addr, float v) {
  if (v >= 0.f) atomicMax((int*)addr, __float_as_int(v));
  else          atomicMin((unsigned int*)addr, __float_as_uint(v));
}

__device__ __forceinline__ float leaky02(float v) { return v > 0.f ? v : 0.2f * v; }

// a_s[n,h] = sum_c H[n,h,c]*a_src[h,c];  a_d likewise. One block per node.
__global__ void k_att_scores(const float* __restrict__ H,
                             const float* __restrict__ a_src,
                             const float* __restrict__ a_dst,
                             float* __restrict__ as_out, float* __restrict__ ad_out,
                             int heads, int C) {
  __shared__ float s1[256], s2[256];
  const int n = blockIdx.x, t = threadIdx.x;
  const int HC = heads * C;
  const float hv = H[(size_t)n * HC + t];
  s1[t] = hv * a_src[t];
  s2[t] = hv * a_dst[t];
  __syncthreads();
  for (int st = C / 2; st > 0; st >>= 1) {
    if ((t & (C - 1)) < st) { s1[t] += s1[t + st]; s2[t] += s2[t + st]; }
    __syncthreads();
  }
  if ((t & (C - 1)) == 0) {
    const int h = t / C;
    as_out[n * heads + h] = s1[t];
    ad_out[n * heads + h] = s2[t];
  }
}

__device__ __forceinline__ void edge_nodes(const int* __restrict__ ei, int E,
                                           int e, int& src, int& dst) {
  if (e < E) { src = ei[e]; dst = ei[E + e]; }
  else       { src = dst = e - E; }
}

__global__ void k_edge_max(const int* __restrict__ ei, int E, int Nn, int heads,
                           const float* __restrict__ as, const float* __restrict__ ad,
                           float* __restrict__ mbuf) {
  const int i = blockIdx.x * blockDim.x + threadIdx.x;
  const int total = (E + Nn) * heads;
  if (i >= total) return;
  const int e = i / heads, h = i - e * heads;
  int src, dst; edge_nodes(ei, E, e, src, dst);
  const float v = leaky02(as[src * heads + h] + ad[dst * heads + h]);
  atomicMaxF(&mbuf[dst * heads + h], v);
}

__global__ void k_edge_sum(const int* __restrict__ ei, int E, int Nn, int heads,
                           const float* __restrict__ as, const float* __restrict__ ad,
                           const float* __restrict__ mbuf, float* __restrict__ sbuf) {
  const int i = blockIdx.x * blockDim.x + threadIdx.x;
  const int total = (E + Nn) * heads;
  if (i >= total) return;
  const int e = i / heads, h = i - e * heads;
  int src, dst; edge_nodes(ei, E, e, src, dst);
  const int di = dst * heads + h;
  const float v = leaky02(as[src * heads + h] + ad[dst * heads + h]);
  atomicAdd(&sbuf[di], expf(v - mbuf[di]));
}

// out[dst,h,c] += H[src,h,c] * alpha(e,h). One block per edge, blockDim = H*C.
__global__ void k_edge_aggr(const int* __restrict__ ei, int E, int Nn, int heads, int C,
                            const float* __restrict__ as, const float* __restrict__ ad,
                            const float* __restrict__ mbuf, const float* __restrict__ sbuf,
                            const float* __restrict__ H, float* __restrict__ out) {
  const int e = blockIdx.x, j = threadIdx.x;
  const int h = j / C;
  int src, dst; edge_nodes(ei, E, e, src, dst);
  const int HC = heads * C;
  // kick off the row gather early; overlaps with the scalar coefficient loads
  __builtin_prefetch(H + (size_t)src * HC + j, 0, 3);
  const int di = dst * heads + h;
  const float v = leaky02(as[src * heads + h] + ad[dst * heads + h]);
  const float alpha = expf(v - mbuf[di]) / (sbuf[di] + 1e-16f);
  atomicAdd(&out[(size_t)dst * HC + j], H[(size_t)src * HC + j] * alpha);
}

// y = elu(h + bias), stored in place; accumulate column sum / sumsq.
// blockDim == D, grid-strided over rows.
__global__ void k_elu_stats(float* __restrict__ hbuf, const float* __restrict__ bias,
                            float* __restrict__ colsum, float* __restrict__ colsumsq,
                            int Nn, int D) {
  const int c = threadIdx.x;
  float s = 0.f, s2 = 0.f;
  for (int r = blockIdx.x; r < Nn; r += gridDim.x) {
    const size_t idx = (size_t)r * D + c;
    float y = hbuf[idx] + bias[c];
    y = y > 0.f ? y : expf(y) - 1.f;
    hbuf[idx] = y;
    s += y; s2 += y * y;
  }
  atomicAdd(&colsum[c], s);
  atomicAdd(&colsumsq[c], s2);
}

__global__ void k_finalize_stats(const float* __restrict__ colsum,
                                 const float* __restrict__ colsumsq,
                                 float* __restrict__ meanb, float* __restrict__ invb,
                                 int Nn, int D) {
  const int c = threadIdx.x;
  if (c >= D) return;
  const float m = colsum[c] / (float)Nn;
  const float v = colsumsq[c] / (float)Nn - m * m;
  meanb[c] = m;
  invb[c]  = rsqrtf(v + 1e-5f);
}

// Fused BN (precomputed mean/inv) then row LayerNorm, in place. Block per row.
__global__ void k_bn_ln(float* __restrict__ hbuf,
                        const float* __restrict__ mean, const float* __restrict__ inv,
                        const float* __restrict__ gbn, const float* __restrict__ bbn,
                        const float* __restrict__ gln, const float* __restrict__ bln,
                        int D) {
  __shared__ float s1[256], s2[256];
  const int r = blockIdx.x, c = threadIdx.x;
  const size_t idx = (size_t)r * D + c;
  const float z = (hbuf[idx] - mean[c]) * inv[c] * gbn[c] + bbn[c];
  s1[c] = z; s2[c] = z * z;
  __syncthreads();
  for (int st = D / 2; st > 0; st >>= 1) {
    if (c < st) { s1[c] += s1[c + st]; s2[c] += s2[c + st]; }
    __syncthreads();
  }
  const float rm = s1[0] / (float)D;
  const float rv = s2[0] / (float)D - rm * rm;
  hbuf[idx] = (z - rm) * rsqrtf(rv + 1e-5f) * gln[c] + bln[c];
}

// Mean pool: block per node, blockDim = 128.
__global__ void k_pool(const float* __restrict__ h, const int* __restrict__ batch_vec,
                       float* __restrict__ gsum, float* __restrict__ gcnt) {
  const int n = blockIdx.x, t = threadIdx.x;
  const int b = batch_vec[n];
  atomicAdd(&gsum[(size_t)b * 128 + t], h[(size_t)n * 128 + t]);
  if (t == 0) atomicAdd(&gcnt[b], 1.f);
}

__global__ void k_pool_div(float* __restrict__ gsum, const float* __restrict__ gcnt,
                           int total) {
  const int i = blockIdx.x * blockDim.x + threadIdx.x;
  if (i >= total) return;
  const float c = gcnt[i / 128];
  gsum[i] /= fmaxf(c, 1.f);
}

// BN over 64 graphs + ReLU, in place. One block of 128 threads (col-per-thread).
__global__ void k_mlp_bn_relu(float* __restrict__ z, const float* __restrict__ g,
                              const float* __restrict__ b, int G) {
  const int c = threadIdx.x;
  float s = 0.f, s2 = 0.f;
  for (int r = 0; r < G; ++r) { const float v = z[r * 128 + c]; s += v; s2 += v * v; }
  const float m  = s / (float)G;
  const float vv = s2 / (float)G - m * m;
  const float iv = rsqrtf(vv + 1e-5f);
  for (int r = 0; r < G; ++r) {
    const float v = (z[r * 128 + c] - m) * iv * g[c] + b[c];
    z[r * 128 + c] = fmaxf(v, 0.f);
  }
}

// out[g,o] = z[g,:] @ W2[:,o] + b2[o]  (64x8)
__global__ void k_mlp_out(const float* __restrict__ z, const float* __restrict__ W2,
                          const float* __restrict__ b2, float* __restrict__ out,
                          int G) {
  const int i = blockIdx.x * blockDim.x + threadIdx.x;
  if (i >= G * 8) return;
  const int gi = i / 8, o = i - gi * 8;
  float acc = b2[o];
  for (int c = 0; c < 128; ++c) acc += z[gi * 128 + c] * W2[c * 8 + o];
  out[i] = acc;
}

// ---------------------------------------------------------------------------
static inline size_t pad64(size_t x) { return (x + 63) & ~(size_t)63; }  // floats

extern "C" void kernel_launch(void* const* d_in, const int* in_sizes, int n_in,
                              void* d_out, int out_size, void* d_ws, size_t ws_size,
                              hipStream_t stream) {
  (void)n_in; (void)ws_size; (void)out_size;
  const float* x    = (const float*)d_in[0];
  const int*   ei   = (const int*)d_in[1];
  const int*   bvec = (const int*)d_in[2];
  // params, setup_inputs() insertion order:
  const float* c1W  = (const float*)d_in[3];
  const float* c1as = (const float*)d_in[4];
  const float* c1ad = (const float*)d_in[5];
  const float* c1b  = (const float*)d_in[6];
  const float* c2W  = (const float*)d_in[7];
  const float* c2as = (const float*)d_in[8];
  const float* c2ad = (const float*)d_in[9];
  const float* c2b  = (const float*)d_in[10];
  const float* c2rW = (const float*)d_in[11];
  const float* c3W  = (const float*)d_in[12];
  const float* c3as = (const float*)d_in[13];
  const float* c3ad = (const float*)d_in[14];
  const float* c3b  = (const float*)d_in[15];
  const float* c3rW = (const float*)d_in[16];
  const float* bn1g = (const float*)d_in[17];
  const float* bn1b = (const float*)d_in[18];
  const float* bn2g = (const float*)d_in[19];
  const float* bn2b = (const float*)d_in[20];
  const float* bn3g = (const float*)d_in[21];
  const float* bn3b = (const float*)d_in[22];
  const float* ln1g = (const float*)d_in[23];
  const float* ln1b = (const float*)d_in[24];
  const float* ln2g = (const float*)d_in[25];
  const float* ln2b = (const float*)d_in[26];
  const float* ln3g = (const float*)d_in[27];
  const float* ln3b = (const float*)d_in[28];
  const float* mW1  = (const float*)d_in[29];
  const float* mb1  = (const float*)d_in[30];
  const float* mbng = (const float*)d_in[31];
  const float* mbnb = (const float*)d_in[32];
  const float* mW2  = (const float*)d_in[33];
  const float* mb2  = (const float*)d_in[34];

  const int Nn = in_sizes[2];          // 50000
  const int E  = in_sizes[1] / 2;      // 800000
  const int ET = E + Nn;               // with self loops
  const int G  = 64;

  // Workspace layout (floats)
  float* base = (float*)d_ws;
  size_t off = 0;
  float* buf0 = base + off; off += pad64((size_t)Nn * 256);  // lin output H
  float* buf1 = base + off; off += pad64((size_t)Nn * 256);  // aggregate / layer state
  float* buf2 = base + off; off += pad64((size_t)Nn * 256);  // aggregate / layer state
  float* as_  = base + off; off += pad64((size_t)Nn * 2);
  float* ad_  = base + off; off += pad64((size_t)Nn * 2);
  float* mbuf = base + off; off += pad64((size_t)Nn * 2);
  float* sbuf = base + off; off += pad64((size_t)Nn * 2);
  float* csum = base + off; off += pad64(256);
  float* csq  = base + off; off += pad64(256);
  float* cmn  = base + off; off += pad64(256);
  float* cinv = base + off; off += pad64(256);
  float* gsum = base + off; off += pad64((size_t)G * 128);
  float* gcnt = base + off; off += pad64(G);
  float* zbuf = base + off; off += pad64((size_t)G * 128);

  const float NEG_INF = -INFINITY;
  auto fill = [&](float* p, float v, size_t n) {
    k_fill<<<(unsigned)((n + 255) / 256), 256, 0, stream>>>(p, v, n);
  };
  auto gemm = [&](const float* A, const float* B, float* C, int M, int N, int K,
                  const float* bias, int acc) {
    k_gemm_wmma<<<dim3(N / 64, M / 16), 32, 0, stream>>>(A, B, C, M, N, K, bias, acc);
  };
  auto gat_layer = [&](const float* in, int K, const float* W, const float* aS,
                       const float* aD, int heads, int C, float* Hlin, float* out) {
    const int HC = heads * C;
    gemm(in, W, Hlin, Nn, HC, K, nullptr, 0);                 // lin: Hlin = in @ W
    k_att_scores<<<Nn, HC, 0, stream>>>(Hlin, aS, aD, as_, ad_, heads, C);
    fill(mbuf, NEG_INF, (size_t)Nn * heads);
    fill(sbuf, 0.f, (size_t)Nn * heads);
    fill(out, 0.f, (size_t)Nn * HC);
    const int tot = ET * heads;
    k_edge_max<<<(tot + 255) / 256, 256, 0, stream>>>(ei, E, Nn, heads, as_, ad_, mbuf);
    k_edge_sum<<<(tot + 255) / 256, 256, 0, stream>>>(ei, E, Nn, heads, as_, ad_, mbuf, sbuf);
    k_edge_aggr<<<ET, HC, 0, stream>>>(ei, E, Nn, heads, C, as_, ad_, mbuf, sbuf, Hlin, out);
  };
  auto norm_block = [&](float* h, const float* bias, int D, const float* gbn,
                        const float* bbn, const float* gln, const float* bln) {
    fill(csum, 0.f, D); fill(csq, 0.f, D);
    k_elu_stats<<<256, D, 0, stream>>>(h, bias, csum, csq, Nn, D);
    k_finalize_stats<<<1, D, 0, stream>>>(csum, csq, cmn, cinv, Nn, D);
    k_bn_ln<<<Nn, D, 0, stream>>>(h, cmn, cinv, gbn, bbn, gln, bln, D);
  };

  // ---- Layer 1: in=x[N,320], heads=2, C=128, no residual ----
  gat_layer(x, 320, c1W, c1as, c1ad, 2, 128, buf0, buf1);
  norm_block(buf1, c1b, 256, bn1g, bn1b, ln1g, ln1b);

  // ---- Layer 2: in=buf1[N,256], heads=2, residual ----
  gat_layer(buf1, 256, c2W, c2as, c2ad, 2, 128, buf0, buf2);
  gemm(buf1, c2rW, buf2, Nn, 256, 256, nullptr, 1);
  norm_block(buf2, c2b, 256, bn2g, bn2b, ln2g, ln2b);

  // ---- Layer 3: in=buf2[N,256], heads=1, C=128, residual ----
  gat_layer(buf2, 256, c3W, c3as, c3ad, 1, 128, buf0, buf1);
  gemm(buf2, c3rW, buf1, Nn, 128, 256, nullptr, 1);
  norm_block(buf1, c3b, 128, bn3g, bn3b, ln3g, ln3b);

  // ---- Global mean pool ----
  fill(gsum, 0.f, (size_t)G * 128);
  fill(gcnt, 0.f, G);
  k_pool<<<Nn, 128, 0, stream>>>(buf1, bvec, gsum, gcnt);
  k_pool_div<<<(G * 128 + 255) / 256, 256, 0, stream>>>(gsum, gcnt, G * 128);

  // ---- MLP head ----
  gemm(gsum, mW1, zbuf, G, 128, 128, mb1, 0);
  k_mlp_bn_relu<<<1, 128, 0, stream>>>(zbuf, mbng, mbnb, G);
  k_mlp_out<<<(G * 8 + 255) / 256, 256, 0, stream>>>(zbuf, mW2, mb2, (float*)d_out, G);
}